// granular_BERT_53678501265786
// MI455X (gfx1250) — compile-verified
//
#include <hip/hip_runtime.h>

typedef float v2f __attribute__((ext_vector_type(2)));
typedef float v4f __attribute__((ext_vector_type(4)));
typedef float v8f __attribute__((ext_vector_type(8)));

#define DDIM 256
#define KCH  (DDIM / 4)       // 64 k-chunks of 4 for 16x16x4 WMMA
#define ROWS_PER_WG 64        // X rows per workgroup (4 waves x 16 rows)
#define TBLK 16               // Y rows per inner step
#define MAX_LY 2048

// out[b, row_off + m, :] = sum_n  1/(1+dist(X[b,m], Y[b,n])) * Y[b,n,:]
// Called twice: (X=context, Y=target, row_off=0)  -> g2c
//              (X=target,  Y=context, row_off=Lc) -> g2t
__global__ __launch_bounds__(128, 1)
void granular_softdist_mm(const float* __restrict__ X,
                          const float* __restrict__ Y,
                          float* __restrict__ out,
                          int Lx, int Ly, int row_off, int out_rows)
{
    __shared__ float ylds[TBLK * DDIM];        // 16 KB: current Y block
    __shared__ float wlds[4 * 16 * 16];        //  4 KB: per-wave weight tile
    __shared__ float yn[MAX_LY];               //  8 KB: |Y row|^2
    __shared__ float xn[ROWS_PER_WG];          // 256 B: |X row|^2

    const int tid  = threadIdx.x;
    const int wid  = tid >> 5;                 // wave id 0..3
    const int lane = tid & 31;
    const int lrow = lane & 15;                // lane % 16
    const int lhi  = lane >> 4;                // 0 or 1
    const int b    = blockIdx.y;

    const size_t xbase = (size_t)b * (size_t)Lx * DDIM;
    const size_t ybase = (size_t)b * (size_t)Ly * DDIM;

    // ---- squared norms into LDS (once per workgroup) ----
    if (tid < ROWS_PER_WG) {
        const int row = blockIdx.x * ROWS_PER_WG + tid;
        const float* p = X + xbase + (size_t)row * DDIM;
        float s = 0.f;
        #pragma unroll 8
        for (int d = 0; d < DDIM; ++d) s += p[d] * p[d];
        xn[tid] = s;
    }
    for (int r = tid; r < Ly; r += 128) {
        const float* p = Y + ybase + (size_t)r * DDIM;
        float s = 0.f;
        #pragma unroll 8
        for (int d = 0; d < DDIM; ++d) s += p[d] * p[d];
        yn[r] = s;
    }

    // ---- preload this wave's 16 X rows as A fragments (registers) ----
    // A layout (16x4 f32): lane L holds (m = L%16, k = (L/16)*2 + j), j = vgpr 0/1
    const int row0 = blockIdx.x * ROWS_PER_WG + wid * 16;
    const float* xrow = X + xbase + (size_t)(row0 + lrow) * DDIM;
    v2f afrag[KCH];
    #pragma unroll
    for (int kc = 0; kc < KCH; ++kc)
        afrag[kc] = *(const v2f*)(xrow + kc * 4 + lhi * 2);

    v8f acc[16];
    #pragma unroll
    for (int f = 0; f < 16; ++f) {
        v8f z = {0.f, 0.f, 0.f, 0.f, 0.f, 0.f, 0.f, 0.f};
        acc[f] = z;
    }

    __syncthreads();   // norms ready

    float* mywl = wlds + wid * 256;

    for (int t0 = 0; t0 < Ly; t0 += TBLK) {
        __syncthreads();   // everyone done reading previous ylds
        // cooperative stage of Y[t0 .. t0+15, 0..255] (4096 floats, 128 thr)
        {
            const float* ysrc = Y + ybase + (size_t)t0 * DDIM;
            #pragma unroll
            for (int c = 0; c < (TBLK * DDIM) / (128 * 4); ++c) {
                const int pos = c * 512 + tid * 4;
                *(v4f*)(ylds + pos) = *(const v4f*)(ysrc + pos);
            }
        }
        __syncthreads();

        // ---- Gram tile: dot[m,n] = X[row0+m,:] . Y[t0+n,:]  (64 WMMAs) ----
        // B layout (4x16 f32): lane L holds (k = (L/16)*2 + j, n = L%16)
        v8f w = {0.f, 0.f, 0.f, 0.f, 0.f, 0.f, 0.f, 0.f};
        #pragma unroll
        for (int kc = 0; kc < KCH; ++kc) {
            v2f bfrag = *(const v2f*)(ylds + lrow * DDIM + kc * 4 + lhi * 2);
            w = __builtin_amdgcn_wmma_f32_16x16x4_f32(
                    false, afrag[kc], false, bfrag, (short)0, w, false, false);
        }

        // ---- elementwise: w = 1/(1 + sqrt(max(c2 + t2 - 2*dot, 0))) ----
        // C/D layout: lane L vgpr g holds (m = g + (L/16)*8, n = L%16)
        const float t2 = yn[t0 + lrow];
        #pragma unroll
        for (int g = 0; g < 8; ++g) {
            const float c2 = xn[wid * 16 + g + lhi * 8];
            float d2 = c2 + t2 - 2.0f * w[g];
            d2 = fmaxf(d2, 0.0f);
            w[g] = __builtin_amdgcn_rcpf(1.0f + __builtin_amdgcn_sqrtf(d2));
        }

        // ---- relayout weight tile D->A via wave-private LDS ----
        #pragma unroll
        for (int g = 0; g < 8; ++g)
            mywl[(g + lhi * 8) * 16 + lrow] = w[g];
        __syncthreads();

        // ---- acc[m, :] += W[m, k] * Y[t0+k, :]  (64 WMMAs) ----
        v2f a2[4];
        #pragma unroll
        for (int kk = 0; kk < 4; ++kk)
            a2[kk] = *(const v2f*)(mywl + lrow * 16 + kk * 4 + lhi * 2);
        #pragma unroll
        for (int f = 0; f < 16; ++f) {
            #pragma unroll
            for (int kk = 0; kk < 4; ++kk) {
                v2f bb;
                bb.x = ylds[(kk * 4 + lhi * 2 + 0) * DDIM + f * 16 + lrow];
                bb.y = ylds[(kk * 4 + lhi * 2 + 1) * DDIM + f * 16 + lrow];
                acc[f] = __builtin_amdgcn_wmma_f32_16x16x4_f32(
                            false, a2[kk], false, bb, (short)0, acc[f], false, false);
            }
        }
    }

    // ---- write the 16x256 output tile of this wave ----
    float* obase = out + ((size_t)b * out_rows + row_off + row0) * DDIM;
    #pragma unroll
    for (int f = 0; f < 16; ++f) {
        #pragma unroll
        for (int g = 0; g < 8; ++g)
            obase[(size_t)(g + lhi * 8) * DDIM + f * 16 + lrow] = acc[f][g];
    }
}

extern "C" void kernel_launch(void* const* d_in, const int* in_sizes, int n_in,
                              void* d_out, int out_size, void* d_ws, size_t ws_size,
                              hipStream_t stream) {
    (void)in_sizes; (void)n_in; (void)d_ws; (void)ws_size; (void)out_size;
    const float* ctx = (const float*)d_in[0];   // [32, 2048, 256]
    const float* tgt = (const float*)d_in[1];   // [32,  512, 256]
    float* out = (float*)d_out;                 // [32, 2560, 256]

    const int B = 32, LC = 2048, LT = 512, OUT_ROWS = LC + LT;

    // g2c: X = context (2048 rows), Y = target (512 rows)
    granular_softdist_mm<<<dim3(LC / ROWS_PER_WG, B), 128, 0, stream>>>(
        ctx, tgt, out, LC, LT, /*row_off=*/0, OUT_ROWS);

    // g2t: X = target (512 rows), Y = context (2048 rows)  (distance symmetric)
    granular_softdist_mm<<<dim3(LT / ROWS_PER_WG, B), 128, 0, stream>>>(
        tgt, ctx, out, LT, LC, /*row_off=*/LC, OUT_ROWS);
}